// VariantCoeLinear1d_83064667505018
// MI455X (gfx1250) — compile-verified
//
#include <hip/hip_runtime.h>
#include <stdint.h>

#define NPTS 2048
#define TPB  256
#define PPT  8   // points per thread: 256*8 = 2048

typedef float v2f __attribute__((ext_vector_type(2)));
typedef float v4f __attribute__((ext_vector_type(4)));

// Pairwise (packed-fp32) flux f(u) and wave speed |f'(u)|:
//   f  = 0.5*u*(3-u^2) + (beta/12)*u^2*(0.75 - 2u + 1.5u^2 - 0.25u^4)
//   f' = 1.5*(1-u^2)   + (beta/12)*(1.5u - 6u^2 + 6u^3 - 1.5u^5)
// Written with 2-wide ext-vectors so the backend can use v_pk_fma_f32 /
// v_pk_mul_f32 / v_pk_add_f32 (VOP3P packed fp32 on CDNA5).
__device__ __forceinline__ void flux_fs_v2(v2f u, v2f& f, v2f& a) {
    const float k = 0.1f / 12.0f;               // beta/12
    v2f u2 = u * u;
    v2f p  = 0.75f + u * (-2.0f + u * (1.5f - 0.25f * u2));
    f = u * (1.5f - 0.5f * u2) + k * (u2 * p);
    v2f q  = 1.5f + u * (-6.0f + u * (6.0f - 1.5f * u2));
    v2f fp = (1.5f - 1.5f * u2) + k * (u * q);
    a = __builtin_elementwise_abs(fp);
}

__global__ void __launch_bounds__(TPB)
llf_timestep_kernel(const float* __restrict__ init,
                    const int* __restrict__ stepnum_p,
                    float* __restrict__ out,
                    int batch)
{
    // ping-pong state buffers in LDS: 2 * 8KB = 16KB per block
    __shared__ __align__(16) float buf[2][NPTS];

    const int t    = threadIdx.x;
    const int b    = blockIdx.x;
    const int base = t * PPT;
    const float c  = (float)(0.002 / (10.0 / 2048.0));   // dt/dx = 0.4096
    const int steps = stepnum_p[0];
    const size_t slice = (size_t)batch * NPTS;

    // ---- load init row into buf[0]; slice 0 of the trajectory is init itself
    const v4f* urow = (const v4f*)(init + (size_t)b * NPTS);
    v4f x0 = urow[t * 2 + 0];
    v4f x1 = urow[t * 2 + 1];
    *(v4f*)&buf[0][base]     = x0;
    *(v4f*)&buf[0][base + 4] = x1;
    v4f* orow0 = (v4f*)(out + (size_t)b * NPTS);
    __builtin_nontemporal_store(x0, &orow0[t * 2 + 0]);
    __builtin_nontemporal_store(x1, &orow0[t * 2 + 1]);

    for (int s = 0; s < steps - 1; ++s) {
        const int cur = s & 1;
        const int nb  = (s + 1) & 1;

        __syncthreads();  // all lanes finished writing buf[cur] last step

        // buf[nb] was async-stored to HBM two steps ago (2 stores/lane/step);
        // allow the most recent step's 2 stores to remain in flight.
        asm volatile("s_wait_asynccnt 0x2" ::: "memory");

        // gather my 8 points + 2 halo points from LDS
        float uu[PPT + 2];
        uu[0] = buf[cur][(t == 0) ? 0 : (base - 1)];
        v4f m0 = *(const v4f*)&buf[cur][base];
        v4f m1 = *(const v4f*)&buf[cur][base + 4];
        uu[1] = m0.x; uu[2] = m0.y; uu[3] = m0.z; uu[4] = m0.w;
        uu[5] = m1.x; uu[6] = m1.y; uu[7] = m1.z; uu[8] = m1.w;
        uu[9] = buf[cur][(t == TPB - 1) ? (NPTS - 1) : (base + PPT)];

        // packed-fp32 flux evaluation, two points per instruction
        float fv[PPT + 2], av[PPT + 2];
        #pragma unroll
        for (int i = 0; i < PPT + 2; i += 2) {
            v2f uv = {uu[i], uu[i + 1]}, fvv, avv;
            flux_fs_v2(uv, fvv, avv);
            fv[i] = fvv.x; fv[i + 1] = fvv.y;
            av[i] = avv.x; av[i + 1] = avv.y;
        }

        // Rusanov interface fluxes (9 interfaces around my 8 points)
        float fh[PPT + 1];
        #pragma unroll
        for (int i = 0; i < PPT + 1; ++i) {
            float alpha = fmaxf(av[i], av[i + 1]);
            fh[i] = 0.5f * ((fv[i] + fv[i + 1]) - alpha * (uu[i + 1] - uu[i]));
        }

        float un[PPT];
        #pragma unroll
        for (int i = 0; i < PPT; ++i)
            un[i] = uu[i + 1] - c * (fh[i + 1] - fh[i]);
        // outflow boundaries: u[0]=u[1], u[N-1]=u[N-2] (post-update values)
        if (t == 0)       un[0] = un[1];
        if (t == TPB - 1) un[PPT - 1] = un[PPT - 2];

        *(v4f*)&buf[nb][base]     = (v4f){un[0], un[1], un[2], un[3]};
        *(v4f*)&buf[nb][base + 4] = (v4f){un[4], un[5], un[6], un[7]};

        // order my ds_stores before the async engine reads them
        asm volatile("s_wait_dscnt 0x0" ::: "memory");

        // CDNA5 async LDS->global streaming store of this step's 32B/lane,
        // non-temporal so the 256MB trajectory doesn't churn L2.
        // (generic LDS pointer: low 32 bits are the LDS byte address)
        unsigned l0 = (unsigned)(uintptr_t)&buf[nb][base];
        unsigned l1 = l0 + 16u;
        uint64_t g0 = (uint64_t)(uintptr_t)(out + (size_t)(s + 1) * slice
                                                + (size_t)b * NPTS + base);
        uint64_t g1 = g0 + 16ull;
        asm volatile("global_store_async_from_lds_b128 %0, %1, off th:TH_STORE_NT"
                     :: "v"(g0), "v"(l0) : "memory");
        asm volatile("global_store_async_from_lds_b128 %0, %1, off th:TH_STORE_NT"
                     :: "v"(g1), "v"(l1) : "memory");
    }
    // s_endpgm performs an implicit wait-idle, draining remaining async stores
}

extern "C" void kernel_launch(void* const* d_in, const int* in_sizes, int n_in,
                              void* d_out, int out_size, void* d_ws, size_t ws_size,
                              hipStream_t stream) {
    const float* init    = (const float*)d_in[0];
    const int*   stepnum = (const int*)d_in[1];
    float*       out     = (float*)d_out;
    int batch = in_sizes[0] / NPTS;   // 128
    llf_timestep_kernel<<<dim3(batch), dim3(TPB), 0, stream>>>(init, stepnum, out, batch);
}